// SlotAttention_26242250179176
// MI455X (gfx1250) — compile-verified
//
#include <hip/hip_runtime.h>
#include <math.h>

typedef float v2f __attribute__((ext_vector_type(2)));
typedef float v8f __attribute__((ext_vector_type(8)));

static constexpr int B_ = 32, S_ = 24, F_ = 4096, DIN_ = 768, D_ = 256, H_ = 1024;
static constexpr int MS_ = B_ * S_; // 768 rows (B*S)

__device__ __forceinline__ v8f wmma4(v2f a, v2f b, v8f c) {
  // D = A(16x4) * B(4x16) + C(16x16), fp32 — v_wmma_f32_16x16x4_f32
  return __builtin_amdgcn_wmma_f32_16x16x4_f32(false, a, false, b, (short)0, c,
                                               false, false);
}

// ---------------------------------------------------------------------------
// Generic GEMM: C[M,N] = act( [LN](A[M,K]) * W[N,K]^T + bias ) [+ R]
// Block = 256 threads (8 wave32s) handling a 16-row M tile; K chunked by 256
// through LDS; each wave owns up to 8 disjoint 16x16 output tiles.
// ---------------------------------------------------------------------------
template <bool LN, bool DUAL, bool RELU, bool RESID>
__global__ void __launch_bounds__(256)
gemm16(const float* __restrict__ A, const float* __restrict__ lng,
       const float* __restrict__ lnb, const float* __restrict__ W0,
       const float* __restrict__ b0, float* C0, const float* __restrict__ W1,
       const float* __restrict__ b1, float* C1, const float* R, int M, int K,
       int N) {
  constexpr int KC = 256;
  __shared__ float tile[16 * (KC + 2)];
  __shared__ float s_mu[16], s_rs[16];
  const int tid = threadIdx.x;
  const int lane = tid & 31;
  const int wv = tid >> 5;
  const int m0 = blockIdx.x * 16;
  const int N16 = N >> 4;
  const int tcount = DUAL ? 2 * N16 : N16;
  const int numT = tcount >> 3; // per wave (8 waves)
  const int nlo = lane & 15;
  const int hi = lane >> 4;

  if (LN) { // full-row stats streamed from global (K can exceed one chunk)
    for (int rr = 0; rr < 2; ++rr) {
      int r = wv * 2 + rr;
      const float* row = A + (size_t)(m0 + r) * K;
      float s = 0.f, sq = 0.f;
      for (int i = lane; i < K; i += 32) {
        float x = row[i];
        s += x;
        sq += x * x;
      }
      for (int off = 16; off > 0; off >>= 1) {
        s += __shfl_xor(s, off, 32);
        sq += __shfl_xor(sq, off, 32);
      }
      if (lane == 0) {
        float mu = s / (float)K;
        s_mu[r] = mu;
        s_rs[r] = rsqrtf(sq / (float)K - mu * mu + 1e-5f);
      }
    }
  }
  __syncthreads();

  v8f acc[8];
#pragma unroll
  for (int i = 0; i < 8; ++i) {
    float bv = 0.f;
    if (i < numT) {
      int t = wv + i * 8;
      int mat = DUAL ? (t >> 4) : 0;
      int nt = DUAL ? (t & 15) : t;
      const float* bias = mat ? b1 : b0;
      if (bias) bv = bias[nt * 16 + nlo];
    }
#pragma unroll
    for (int v = 0; v < 8; ++v) acc[i][v] = bv;
  }

  for (int kc = 0; kc < K; kc += KC) {
    for (int idx = tid; idx < 16 * KC; idx += 256) {
      int r = idx >> 8;
      int c = idx & (KC - 1);
      float v = A[(size_t)(m0 + r) * K + kc + c];
      if (LN) v = (v - s_mu[r]) * s_rs[r] * lng[kc + c] + lnb[kc + c];
      tile[r * (KC + 2) + c] = v;
    }
    __syncthreads();
    for (int kk = 0; kk < KC; kk += 4) {
      int k0 = kk + hi * 2;
      v2f a = *(const v2f*)&tile[nlo * (KC + 2) + k0];
      int gk = kc + k0;
#pragma unroll
      for (int i = 0; i < 8; ++i) {
        if (i < numT) {
          int t = wv + i * 8;
          int mat = DUAL ? (t >> 4) : 0;
          int nt = DUAL ? (t & 15) : t;
          const float* W = mat ? W1 : W0;
          v2f bb = *(const v2f*)&W[(size_t)(nt * 16 + nlo) * K + gk];
          acc[i] = wmma4(a, bb, acc[i]);
        }
      }
    }
    __syncthreads();
  }

#pragma unroll
  for (int i = 0; i < 8; ++i) {
    if (i < numT) {
      int t = wv + i * 8;
      int mat = DUAL ? (t >> 4) : 0;
      int nt = DUAL ? (t & 15) : t;
      float* C = mat ? C1 : C0;
      int n = nt * 16 + nlo;
#pragma unroll
      for (int v = 0; v < 8; ++v) {
        int row = hi * 8 + v;
        float val = acc[i][v];
        if (RELU) val = fmaxf(val, 0.f);
        if (RESID) val += R[(size_t)(m0 + row) * N + n];
        C[(size_t)(m0 + row) * N + n] = val;
      }
    }
  }
}

// ---------------------------------------------------------------------------
// Row LayerNorm for D=256 rows (one block per row, one element per thread).
// ---------------------------------------------------------------------------
__global__ void __launch_bounds__(256)
rowln(const float* __restrict__ X, const float* __restrict__ g,
      const float* __restrict__ b, float* __restrict__ Y) {
  __shared__ float red[8], red2[8];
  __shared__ float s_mu, s_rs;
  int row = blockIdx.x, tid = threadIdx.x;
  float x = X[(size_t)row * D_ + tid];
  float s = x, sq = x * x;
  for (int off = 16; off > 0; off >>= 1) {
    s += __shfl_xor(s, off, 32);
    sq += __shfl_xor(sq, off, 32);
  }
  if ((tid & 31) == 0) {
    red[tid >> 5] = s;
    red2[tid >> 5] = sq;
  }
  __syncthreads();
  if (tid == 0) {
    float S = 0.f, Q = 0.f;
    for (int i = 0; i < 8; ++i) {
      S += red[i];
      Q += red2[i];
    }
    float mu = S / (float)D_;
    s_mu = mu;
    s_rs = rsqrtf(Q / (float)D_ - mu * mu + 1e-5f);
  }
  __syncthreads();
  Y[(size_t)row * D_ + tid] = (x - s_mu) * s_rs * g[tid] + b[tid];
}

// ---------------------------------------------------------------------------
// dots = scale * q[b] @ keys[b]^T, then softmax over the slot axis (24 rows),
// writing pre-norm attention straight into d_out's attn region.
// Block: (fblock of 128 cols, batch b). 8 waves: wave w -> Mtile=w&1,
// Ntiles {w>>1, (w>>1)+4}.
// ---------------------------------------------------------------------------
__global__ void __launch_bounds__(256)
dots_softmax(const float* __restrict__ q, const float* __restrict__ keys,
             float* __restrict__ attn, float scale) {
  __shared__ float qs[32 * 258];
  __shared__ float dots[32 * 132];
  const int b = blockIdx.y;
  const int f0 = blockIdx.x * 128;
  const int tid = threadIdx.x, lane = tid & 31, wv = tid >> 5;
  const int nlo = lane & 15, hi = lane >> 4;

  for (int idx = tid; idx < 32 * 256; idx += 256) {
    int r = idx >> 8, c = idx & 255;
    qs[r * 258 + c] = (r < S_) ? q[((size_t)b * S_ + r) * D_ + c] : 0.f;
  }
  __syncthreads();

  const int Mt = wv & 1;
  const int Nt0 = wv >> 1;
  const int m = Mt * 16 + nlo;
  v8f acc0 = {}, acc1 = {};
  const float* krow0 = keys + ((size_t)b * F_ + f0 + Nt0 * 16 + nlo) * D_;
  const float* krow1 = keys + ((size_t)b * F_ + f0 + (Nt0 + 4) * 16 + nlo) * D_;
  for (int kk = 0; kk < D_; kk += 4) {
    int k0 = kk + hi * 2;
    v2f a = *(const v2f*)&qs[m * 258 + k0];
    v2f bb0 = *(const v2f*)&krow0[k0];
    v2f bb1 = *(const v2f*)&krow1[k0];
    acc0 = wmma4(a, bb0, acc0);
    acc1 = wmma4(a, bb1, acc1);
  }
#pragma unroll
  for (int v = 0; v < 8; ++v) {
    int row = Mt * 16 + hi * 8 + v;
    dots[row * 132 + Nt0 * 16 + nlo] = acc0[v] * scale;
    dots[row * 132 + (Nt0 + 4) * 16 + nlo] = acc1[v] * scale;
  }
  __syncthreads();
  if (tid < 128) {
    int col = tid;
    float mx = -1e30f;
    for (int s = 0; s < S_; ++s) mx = fmaxf(mx, dots[s * 132 + col]);
    float e[S_];
    float sum = 0.f;
    for (int s = 0; s < S_; ++s) {
      e[s] = expf(dots[s * 132 + col] - mx);
      sum += e[s];
    }
    float inv = 1.f / sum;
    for (int s = 0; s < S_; ++s)
      attn[((size_t)b * S_ + s) * F_ + f0 + col] = e[s] * inv;
  }
}

// Deterministic per-(b,s) sum over F of (pre_norm + EPS) -> 1/sum.
__global__ void __launch_bounds__(256)
attn_rowsum(const float* __restrict__ attn, float* __restrict__ invsum) {
  __shared__ float red[8];
  int row = blockIdx.x, tid = threadIdx.x;
  const float* p = attn + (size_t)row * F_;
  float s = 0.f;
  for (int i = tid; i < F_; i += 256) s += p[i];
  for (int off = 16; off > 0; off >>= 1) s += __shfl_xor(s, off, 32);
  if ((tid & 31) == 0) red[tid >> 5] = s;
  __syncthreads();
  if (tid == 0) {
    float S = 0.f;
    for (int i = 0; i < 8; ++i) S += red[i];
    invsum[row] = 1.f / (S + (float)F_ * 1e-8f);
  }
}

// updates[b] = diag(invsum) * (attn[b]+EPS) @ values[b].  K = F = 4096,
// chunked 128-wide through LDS. Block: (dchunk of 64 cols, batch b);
// 8 waves = 2 Mtiles x 4 Ntiles, one 16x16 tile per wave.
__global__ void __launch_bounds__(256)
attn_updates(const float* __restrict__ attn, const float* __restrict__ values,
             const float* __restrict__ invsum, float* __restrict__ updates) {
  __shared__ float at[32 * 130];
  const int b = blockIdx.y;
  const int d0 = blockIdx.x * 64;
  const int tid = threadIdx.x, lane = tid & 31, wv = tid >> 5;
  const int nlo = lane & 15, hi = lane >> 4;
  const int Mt = wv & 1, Nt = wv >> 1;
  const int m = Mt * 16 + nlo;
  const int n = d0 + Nt * 16 + nlo;
  v8f acc = {};
  for (int kc = 0; kc < F_; kc += 128) {
    for (int idx = tid; idx < 32 * 128; idx += 256) {
      int r = idx >> 7, c = idx & 127;
      at[r * 130 + c] =
          (r < S_) ? attn[((size_t)b * S_ + r) * F_ + kc + c] + 1e-8f : 0.f;
    }
    __syncthreads();
    for (int kk = 0; kk < 128; kk += 4) {
      int k0 = kk + hi * 2;
      v2f a = *(const v2f*)&at[m * 130 + k0];
      int gk = kc + k0;
      v2f bb;
      bb.x = values[((size_t)b * F_ + gk) * D_ + n];
      bb.y = values[((size_t)b * F_ + gk + 1) * D_ + n];
      acc = wmma4(a, bb, acc);
    }
    __syncthreads();
  }
#pragma unroll
  for (int v = 0; v < 8; ++v) {
    int row = Mt * 16 + hi * 8 + v;
    if (row < S_)
      updates[((size_t)b * S_ + row) * D_ + n] = acc[v] * invsum[b * S_ + row];
  }
}

// GRU gates: slots = (1-z)*tanh(i_n + r*h_n) + z*s_n
__global__ void __launch_bounds__(256)
gru_gate(const float* __restrict__ gi, const float* __restrict__ gh,
         const float* __restrict__ sn, float* __restrict__ slots) {
  int row = blockIdx.x, d = threadIdx.x;
  size_t o3 = (size_t)row * (3 * D_);
  size_t o1 = (size_t)row * D_ + d;
  float ir = gi[o3 + d], iz = gi[o3 + D_ + d], in = gi[o3 + 2 * D_ + d];
  float hr = gh[o3 + d], hz = gh[o3 + D_ + d], hn = gh[o3 + 2 * D_ + d];
  float r = 1.f / (1.f + expf(-(ir + hr)));
  float z = 1.f / (1.f + expf(-(iz + hz)));
  float nn = tanhf(in + r * hn);
  slots[o1] = (1.f - z) * nn + z * sn[o1];
}

extern "C" void kernel_launch(void* const* d_in, const int* in_sizes, int n_in,
                              void* d_out, int out_size, void* d_ws,
                              size_t ws_size, hipStream_t stream) {
  const float* slots_in = (const float*)d_in[0];
  const float* features = (const float*)d_in[1];
  const float* Wk = (const float*)d_in[2];
  const float* Wv = (const float*)d_in[3];
  const float* Wq = (const float*)d_in[4];
  const float* w_ih = (const float*)d_in[5];
  const float* w_hh = (const float*)d_in[6];
  const float* b_ih = (const float*)d_in[7];
  const float* b_hh = (const float*)d_in[8];
  const float* lf_g = (const float*)d_in[9];
  const float* lf_b = (const float*)d_in[10];
  const float* ls_g = (const float*)d_in[11];
  const float* ls_b = (const float*)d_in[12];
  const float* ml_g = (const float*)d_in[13];
  const float* ml_b = (const float*)d_in[14];
  const float* w1 = (const float*)d_in[15];
  const float* b1 = (const float*)d_in[16];
  const float* w2 = (const float*)d_in[17];
  const float* b2 = (const float*)d_in[18];

  float* ws = (float*)d_ws;
  size_t off = 0;
  float* keys = ws + off;   off += (size_t)B_ * F_ * D_;   // 33.5M
  float* values = ws + off; off += (size_t)B_ * F_ * D_;   // 33.5M
  float* s_n = ws + off;    off += (size_t)MS_ * D_;
  float* q = ws + off;      off += (size_t)MS_ * D_;
  float* upd = ws + off;    off += (size_t)MS_ * D_;
  float* slots = ws + off;  off += (size_t)MS_ * D_;
  float* tbuf = ws + off;   off += (size_t)MS_ * D_;
  float* gi = ws + off;     off += (size_t)MS_ * 3 * D_;
  float* gh = ws + off;     off += (size_t)MS_ * 3 * D_;
  float* hbuf = ws + off;   off += (size_t)MS_ * H_;
  float* invs = ws + off;   off += (size_t)MS_;

  float* out_slots = (float*)d_out;
  float* out_attn = out_slots + (size_t)MS_ * D_;

  hipMemcpyAsync(slots, slots_in, (size_t)MS_ * D_ * sizeof(float),
                 hipMemcpyDeviceToDevice, stream);

  // keys/values: LN(features) @ {Wk,Wv}^T fused (one pass over 402MB input)
  gemm16<true, true, false, false><<<(B_ * F_) / 16, 256, 0, stream>>>(
      features, lf_g, lf_b, Wk, nullptr, keys, Wv, nullptr, values, nullptr,
      B_ * F_, DIN_, D_);

  const float scale = 0.0625f; // 256^-0.5
  for (int it = 0; it < 3; ++it) {
    rowln<<<MS_, 256, 0, stream>>>(slots, ls_g, ls_b, s_n);
    gemm16<false, false, false, false><<<MS_ / 16, 256, 0, stream>>>(
        s_n, nullptr, nullptr, Wq, nullptr, q, nullptr, nullptr, nullptr,
        nullptr, MS_, D_, D_);
    dots_softmax<<<dim3(F_ / 128, B_), 256, 0, stream>>>(q, keys, out_attn,
                                                         scale);
    attn_rowsum<<<MS_, 256, 0, stream>>>(out_attn, invs);
    attn_updates<<<dim3(D_ / 64, B_), 256, 0, stream>>>(out_attn, values, invs,
                                                        upd);
    gemm16<false, false, false, false><<<MS_ / 16, 256, 0, stream>>>(
        upd, nullptr, nullptr, w_ih, b_ih, gi, nullptr, nullptr, nullptr,
        nullptr, MS_, D_, 3 * D_);
    gemm16<false, false, false, false><<<MS_ / 16, 256, 0, stream>>>(
        s_n, nullptr, nullptr, w_hh, b_hh, gh, nullptr, nullptr, nullptr,
        nullptr, MS_, D_, 3 * D_);
    gru_gate<<<MS_, 256, 0, stream>>>(gi, gh, s_n, slots);
    rowln<<<MS_, 256, 0, stream>>>(slots, ml_g, ml_b, tbuf);
    gemm16<false, false, true, false><<<MS_ / 16, 256, 0, stream>>>(
        tbuf, nullptr, nullptr, w1, b1, hbuf, nullptr, nullptr, nullptr,
        nullptr, MS_, D_, H_);
    gemm16<false, false, false, true><<<MS_ / 16, 256, 0, stream>>>(
        hbuf, nullptr, nullptr, w2, b2, slots, nullptr, nullptr, nullptr, slots,
        MS_, H_, D_);
  }
  hipMemcpyAsync(out_slots, slots, (size_t)MS_ * D_ * sizeof(float),
                 hipMemcpyDeviceToDevice, stream);
}